// Matcher_60172491817141
// MI455X (gfx1250) — compile-verified
//
#include <hip/hip_runtime.h>
#include <hip/hip_bf16.h>
#include <math.h>

// Problem constants (from reference)
#define BSZ 32
#define NQ  256
#define NT  512
#define INFV 1e30f

typedef __attribute__((ext_vector_type(2))) float v2f;
typedef __attribute__((ext_vector_type(8))) float v8f;

// ---------------------------------------------------------------------------
// Kernel 1: cost matrix.
// One wave per 16x16 output tile. The rotation inner product (K=4) is done
// with V_WMMA_F32_16X16X4_F32; the 19-dim L1 part is VALU. Memory bound on
// the 16.8MB cost-matrix store, so this is ~1us territory at 23.3 TB/s.
// ---------------------------------------------------------------------------
__global__ void __launch_bounds__(256)
cost_kernel(const float* __restrict__ trans,   // [B, NQ, 3]
            const float* __restrict__ qpos,    // [B, NQ, 16]
            const float* __restrict__ rot,     // [B, NQ, 4]
            const float* __restrict__ pose,    // [B, NT, 23]
            float* __restrict__ out)           // [B, NQ, NT]
{
    const int wavesPerBlock = blockDim.x >> 5;               // 8
    const int w    = threadIdx.x >> 5;
    const int lane = threadIdx.x & 31;
    const int gw   = blockIdx.x * wavesPerBlock + w;         // 0..16383

    const int b    = gw >> 9;                                // / (16*32)
    const int rem  = gw & 511;
    const int M0   = (rem >> 5) << 4;                        // query tile base
    const int N0   = (rem & 31) << 4;                        // target tile base

    const int half = lane >> 4;                              // 0: K0/K1 (lanes 0-15), 1: K2/K3
    const int lm   = lane & 15;

    // ---- A fragment: rotation rows, 16x4 f32 (A layout: lane<16 holds K0,K1; lane>=16 holds K2,K3)
    const float* arow = rot + ((size_t)b * NQ + (M0 + lm)) * 4;
    v2f a;
    a.x = arow[2 * half + 0];
    a.y = arow[2 * half + 1];

    // ---- B fragment: pose quaternion columns, 4x16 f32 (mirror layout over N)
    const float* prow = pose + ((size_t)b * NT + (N0 + lm)) * 23;
    v2f bm;
    bm.x = prow[19 + 2 * half + 0];
    bm.y = prow[19 + 2 * half + 1];

    v8f c = {};
    // D = A x B (+0): per-lane d[r] -> element (M0 + r + 8*half, N0 + lm)
    v8f dot = __builtin_amdgcn_wmma_f32_16x16x4_f32(
        /*neg_a=*/false, a, /*neg_b=*/false, bm,
        /*c_mod=*/(short)0, c, /*reuse_a=*/false, /*reuse_b=*/false);

    // ---- per-lane target vector (column N = N0 + lm) and validity mask
    const int N = N0 + lm;
    float tv[19];
    float vsum = 0.f;
#pragma unroll
    for (int d = 0; d < 19; ++d) { tv[d] = prow[d]; vsum += fabsf(tv[d]); }
#pragma unroll
    for (int d = 19; d < 23; ++d) vsum += fabsf(prow[d]);
    const bool valid = vsum > 0.f;

#pragma unroll
    for (int r = 0; r < 8; ++r) {
        const int M = M0 + r + (half << 3);
        const float* t3  = trans + ((size_t)b * NQ + M) * 3;
        const float* q16 = qpos  + ((size_t)b * NQ + M) * 16;
        float acc = fabsf(t3[0] - tv[0]) + fabsf(t3[1] - tv[1]) + fabsf(t3[2] - tv[2]);
#pragma unroll
        for (int d = 0; d < 16; ++d) acc += fabsf(q16[d] - tv[d + 3]);
        const float cv = acc + 1.0f - fabsf(dot[r]);
        out[((size_t)b * NQ + M) * NT + N] = valid ? cv : INFV;
    }
}

// ---------------------------------------------------------------------------
// Kernel 2: batched Jonker-Volgenant LSAP. One workgroup per batch, one
// thread per column (512). Latency bound; cost rows are L2-resident.
// Argmin: wave32 shuffle reduction + 16-entry cross-wave LDS pass.
// ---------------------------------------------------------------------------
__global__ void __launch_bounds__(NT)
hungarian_kernel(const float* __restrict__ cost_all,  // [B, NQ, NT]
                 float* __restrict__ inds_out,        // [B, NQ]
                 float* __restrict__ mask_out)        // [B, NQ]
{
    const int b   = blockIdx.x;
    const int tid = threadIdx.x;          // 0..511 (column id)
    const float* cost = cost_all + (size_t)b * NQ * NT;

    __shared__ float u[NQ], v[NT], spc[NT];
    __shared__ int   path[NT], col4row[NQ], row4col[NT];
    __shared__ int   SR[NQ], SC[NT];
    __shared__ float wval[16];
    __shared__ int   widx[16];
    __shared__ int   s_i, s_sink;
    __shared__ float s_minVal;

    if (tid < NQ) { u[tid] = 0.f; col4row[tid] = -1; }
    v[tid] = 0.f; row4col[tid] = -1;
    __syncthreads();

    for (int cur = 0; cur < NQ; ++cur) {
        if (tid < NQ) SR[tid] = 0;
        SC[tid] = 0; spc[tid] = INFV; path[tid] = -1;
        if (tid == 0) { s_i = cur; s_minVal = 0.f; s_sink = -1; }
        __syncthreads();

        // ---- shortest augmenting path search ----
        while (true) {
            const int   i      = s_i;
            const float minVal = s_minVal;
            if (tid == 0) SR[i] = 1;

            const float r = minVal + cost[(size_t)i * NT + tid] - u[i] - v[tid];
            if (!SC[tid] && r < spc[tid]) { spc[tid] = r; path[tid] = i; }

            float bv = SC[tid] ? INFV : spc[tid];
            int   bi = tid;
            // wave32 argmin (prefer lower index on ties, matching jnp.argmin)
#pragma unroll
            for (int off = 16; off > 0; off >>= 1) {
                float ov = __shfl_xor(bv, off, 32);
                int   oi = __shfl_xor(bi, off, 32);
                if (ov < bv || (ov == bv && oi < bi)) { bv = ov; bi = oi; }
            }
            if ((tid & 31) == 0) { wval[tid >> 5] = bv; widx[tid >> 5] = bi; }
            __syncthreads();

            if (tid == 0) {
                float best = wval[0]; int bj = widx[0];
#pragma unroll
                for (int k = 1; k < 16; ++k)
                    if (wval[k] < best || (wval[k] == best && widx[k] < bj)) {
                        best = wval[k]; bj = widx[k];
                    }
                s_minVal = best;
                SC[bj] = 1;
                const int r4 = row4col[bj];
                if (r4 < 0) s_sink = bj; else s_i = r4;
            }
            __syncthreads();
            if (s_sink >= 0) break;
        }

        // ---- dual variable updates (use pre-augmentation col4row) ----
        const float minF = s_minVal;
        if (tid < NQ && SR[tid]) {
            float du;
            if (tid == cur) du = minF;
            else {
                const int c4 = col4row[tid];
                du = (c4 >= 0) ? (minF - spc[c4]) : minF;
            }
            u[tid] += du;
        }
        if (SC[tid]) v[tid] -= (minF - spc[tid]);
        __syncthreads();

        // ---- augment along alternating path ----
        if (tid == 0) {
            int j = s_sink;
            while (true) {
                const int i  = path[j];
                row4col[j]   = i;
                const int nj = col4row[i];
                col4row[i]   = j;
                if (i == cur) break;
                j = nj;
            }
        }
        __syncthreads();
    }

    if (tid < NQ) {
        inds_out[(size_t)b * NQ + tid] = (float)col4row[tid];
        mask_out[(size_t)b * NQ + tid] = 1.0f;
    }
}

// ---------------------------------------------------------------------------
extern "C" void kernel_launch(void* const* d_in, const int* in_sizes, int n_in,
                              void* d_out, int out_size, void* d_ws, size_t ws_size,
                              hipStream_t stream) {
    const float* trans = (const float*)d_in[0];   // [B, NQ, 3]
    const float* qpos  = (const float*)d_in[1];   // [B, NQ, 16]
    const float* rot   = (const float*)d_in[2];   // [B, NQ, 4]
    const float* pose  = (const float*)d_in[3];   // [B, NT, 23]

    float* out  = (float*)d_out;
    float* costp = out;                                    // B*NQ*NT
    float* inds  = out + (size_t)BSZ * NQ * NT;            // B*NQ
    float* mask  = inds + (size_t)BSZ * NQ;                // B*NQ

    // 16384 tiles of 16x16, 8 waves (256 threads) per block -> 2048 blocks
    cost_kernel<<<2048, 256, 0, stream>>>(trans, qpos, rot, pose, costp);
    // one workgroup per batch, one thread per column
    hungarian_kernel<<<BSZ, NT, 0, stream>>>(costp, inds, mask);
}